// BCELoss_43654047597080
// MI455X (gfx1250) — compile-verified
//
#include <hip/hip_runtime.h>
#include <stdint.h>

// MI455X / gfx1250, wave32. Radix-select top-k BCE (OHEM) loss.
//
// Fast path (ws_size >= 40KB + 4*n): pass1 streams the 354MB inputs once,
// computes stats + level-0 histogram, and writes a dense 118MB negative-loss
// cache (sentinel -1.0f for non-negatives). 118MB < 192MB global L2, so the
// three refinement/sum passes over the cache are L2-resident. HBM traffic
// ~472MB (~20us @ 23.3TB/s) vs 1.42GB for the recompute fallback.

#define NBINS 2048
#define GRIDB 2048
#define BLOCK 256

typedef __attribute__((ext_vector_type(2))) float v2f;
typedef __attribute__((ext_vector_type(8))) float v8f;

struct SelState {
  unsigned int pos_count;  // atomic
  unsigned int neg_count;  // atomic
  unsigned int k;          // min(neg, floor(3*pos))
  unsigned int c;          // count strictly above current prefix subtree
  unsigned int prefix;     // growing prefix; after level 2 == threshold bits T
  unsigned int krem;       // k - c after final level (# copies of T to take)
  float pos_sum;
  unsigned int pad;
};

// ws layout (bytes):
//   [0,64)            SelState
//   [64, +8192)       hist0 (2048 u32)  bits[31:21]
//   [8256, +8192)     hist1 (2048 u32)  bits[20:10]
//   [16448, +4096)    hist2 (1024 u32)  bits[9:0]
//   [20544, +8192)    partials1 (GRIDB floats)
//   [28736, +8192)    partials4 (GRIDB floats)
//   [40960, +4*n)     vcache (fast path only)

__device__ __forceinline__ float bce_logits(float x, float y) {
  // Explicit fma so codegen (and the bit pattern of v) is identical in every
  // pass — required for the cross-pass radix select in the fallback path.
  float t = __fmaf_rn(-x, y, fmaxf(x, 0.0f));
  return t + log1pf(expf(-fabsf(x)));
}

// Wave32 sum reduction via V_WMMA_F32_16X16X4_F32 (fp32 A/B, full precision).
// A: VGPR0 = lane value, VGPR1 = 0  ->  A[m][0]=x_m, A[m][2]=x_{m+16}
// B: all ones (4x16)                ->  D[m][n] = x_m + x_{m+16}
__device__ __forceinline__ float wave_sum_wmma(float x) {
  v2f a; a.x = x; a.y = 0.0f;
  v2f b; b.x = 1.0f; b.y = 1.0f;
  v8f c = {0.f, 0.f, 0.f, 0.f, 0.f, 0.f, 0.f, 0.f};
  v8f d = __builtin_amdgcn_wmma_f32_16x16x4_f32(
      false, a, false, b, (short)0, c, false, false);
  float s = d[0] + d[1] + d[2] + d[3] + d[4] + d[5] + d[6] + d[7];
  int lo = __builtin_amdgcn_readlane(__float_as_int(s), 0);
  int hi = __builtin_amdgcn_readlane(__float_as_int(s), 16);
  return __int_as_float(lo) + __int_as_float(hi);
}

__global__ void k_init(unsigned int* p, int n) {
  int i = blockIdx.x * blockDim.x + threadIdx.x;
  for (; i < n; i += gridDim.x * blockDim.x) p[i] = 0u;
}

// Pass 1: counts, pos-loss sum, level-0 histogram; optionally write loss cache.
template <int CACHE>
__global__ void k_pass1(const float4* __restrict__ pred, const float4* __restrict__ gt,
                        const float4* __restrict__ mask, int n4,
                        SelState* st, unsigned int* hist0, float* partials,
                        float4* __restrict__ vcache) {
  __shared__ unsigned int lh[NBINS];
  __shared__ unsigned int lcnt[2];
  __shared__ float lsum[BLOCK / 32];
  for (int i = threadIdx.x; i < NBINS; i += BLOCK) lh[i] = 0u;
  if (threadIdx.x < 2) lcnt[threadIdx.x] = 0u;
  __syncthreads();

  unsigned int cp = 0, cn = 0;
  float psum = 0.f;
  const int stride = GRIDB * BLOCK;
  for (int i = blockIdx.x * BLOCK + threadIdx.x; i < n4; i += stride) {
    float4 p = pred[i]; float4 g = gt[i]; float4 m = mask[i];
    if (i + stride < n4) {
      __builtin_prefetch((const void*)(pred + i + stride), 0, 1);
      __builtin_prefetch((const void*)(gt + i + stride), 0, 1);
      __builtin_prefetch((const void*)(mask + i + stride), 0, 1);
    }
    float px[4] = {p.x, p.y, p.z, p.w};
    float gx[4] = {g.x, g.y, g.z, g.w};
    float mx[4] = {m.x, m.y, m.z, m.w};
    float cx[4];
#pragma unroll
    for (int j = 0; j < 4; ++j) {
      float x = px[j], y = gx[j], mm = mx[j];
      float v = bce_logits(x, y);
      bool neg = ((1.0f - y) * mm) > 0.0f;
      if ((y * mm) > 0.0f) { cp++; psum += v; }
      if (neg) {
        cn++;
        atomicAdd(&lh[__float_as_uint(v) >> 21], 1u);
      }
      cx[j] = neg ? v : -1.0f;  // sentinel: sign bit set, never matches a prefix
    }
    if (CACHE) {
      float4 o; o.x = cx[0]; o.y = cx[1]; o.z = cx[2]; o.w = cx[3];
      vcache[i] = o;
    }
  }
  atomicAdd(&lcnt[0], cp);
  atomicAdd(&lcnt[1], cn);

  float wsum = wave_sum_wmma(psum);  // full EXEC here (no divergence)
  int lane = threadIdx.x & 31, wid = threadIdx.x >> 5;
  if (lane == 0) lsum[wid] = wsum;
  __syncthreads();
  if (threadIdx.x == 0) {
    float bs = 0.f;
#pragma unroll
    for (int w = 0; w < BLOCK / 32; ++w) bs += lsum[w];
    partials[blockIdx.x] = bs;
    atomicAdd(&st->pos_count, lcnt[0]);
    atomicAdd(&st->neg_count, lcnt[1]);
  }
  for (int b = threadIdx.x; b < NBINS; b += BLOCK) {
    unsigned int h = lh[b];
    if (h) atomicAdd(&hist0[b], h);
  }
}

__global__ void k_select0(const float* __restrict__ partials, SelState* st,
                          const unsigned int* __restrict__ hist0) {
  __shared__ float red[256];
  float s = 0.f;
  for (int i = threadIdx.x; i < GRIDB; i += 256) s += partials[i];
  red[threadIdx.x] = s;
  __syncthreads();
  for (int off = 128; off > 0; off >>= 1) {
    if (threadIdx.x < off) red[threadIdx.x] += red[threadIdx.x + off];
    __syncthreads();
  }
  if (threadIdx.x == 0) {
    st->pos_sum = red[0];
    unsigned int pos = st->pos_count, neg = st->neg_count;
    // same fp32 math as reference: floor(float(pos) * 3.0)
    unsigned int kf = (unsigned int)floorf((float)pos * 3.0f);
    unsigned int k = (neg < kf) ? neg : kf;
    st->k = k;
    unsigned int acc = 0, chosen = 0, cab = 0;
    bool found = false;
    for (int b = NBINS - 1; b >= 0; --b) {
      unsigned int h = hist0[b];
      if (acc + h >= k) { chosen = (unsigned int)b; cab = acc; found = true; break; }
      acc += h;
    }
    if (!found) { chosen = 0; cab = acc; }
    st->prefix = chosen;  // 11 bits
    st->c = cab;
  }
}

// ---------- fallback path: recompute loss from inputs ----------
template <int LEVEL>
__global__ void k_hist_pass(const float4* __restrict__ pred, const float4* __restrict__ gt,
                            const float4* __restrict__ mask, int n4,
                            const SelState* __restrict__ st, unsigned int* hist) {
  __shared__ unsigned int lh[NBINS];
  const int nb = (LEVEL == 1) ? NBINS : 1024;
  for (int i = threadIdx.x; i < nb; i += BLOCK) lh[i] = 0u;
  __syncthreads();
  const unsigned int prefix = st->prefix;
  const int stride = GRIDB * BLOCK;
  for (int i = blockIdx.x * BLOCK + threadIdx.x; i < n4; i += stride) {
    float4 p = pred[i]; float4 g = gt[i]; float4 m = mask[i];
    float px[4] = {p.x, p.y, p.z, p.w};
    float gx[4] = {g.x, g.y, g.z, g.w};
    float mx[4] = {m.x, m.y, m.z, m.w};
#pragma unroll
    for (int j = 0; j < 4; ++j) {
      float x = px[j], y = gx[j], mm = mx[j];
      if (((1.0f - y) * mm) > 0.0f) {
        unsigned int u = __float_as_uint(bce_logits(x, y));
        if (LEVEL == 1) {
          if ((u >> 21) == prefix) atomicAdd(&lh[(u >> 10) & 0x7FFu], 1u);
        } else {
          if ((u >> 10) == prefix) atomicAdd(&lh[u & 0x3FFu], 1u);
        }
      }
    }
  }
  __syncthreads();
  for (int b = threadIdx.x; b < nb; b += BLOCK) {
    unsigned int h = lh[b];
    if (h) atomicAdd(&hist[b], h);
  }
}

__global__ void k_pass4(const float4* __restrict__ pred, const float4* __restrict__ gt,
                        const float4* __restrict__ mask, int n4,
                        const SelState* __restrict__ st, float* partials) {
  __shared__ float lsum[BLOCK / 32];
  const unsigned int T = st->prefix;
  float s = 0.f;
  const int stride = GRIDB * BLOCK;
  for (int i = blockIdx.x * BLOCK + threadIdx.x; i < n4; i += stride) {
    float4 p = pred[i]; float4 g = gt[i]; float4 m = mask[i];
    float px[4] = {p.x, p.y, p.z, p.w};
    float gx[4] = {g.x, g.y, g.z, g.w};
    float mx[4] = {m.x, m.y, m.z, m.w};
#pragma unroll
    for (int j = 0; j < 4; ++j) {
      float x = px[j], y = gx[j], mm = mx[j];
      if (((1.0f - y) * mm) > 0.0f) {
        float v = bce_logits(x, y);
        if (__float_as_uint(v) > T) s += v;
      }
    }
  }
  float wsum = wave_sum_wmma(s);
  int lane = threadIdx.x & 31, wid = threadIdx.x >> 5;
  if (lane == 0) lsum[wid] = wsum;
  __syncthreads();
  if (threadIdx.x == 0) {
    float bs = 0.f;
#pragma unroll
    for (int w = 0; w < BLOCK / 32; ++w) bs += lsum[w];
    partials[blockIdx.x] = bs;
  }
}

// ---------- fast path: read the (L2-resident) loss cache ----------
template <int LEVEL>
__global__ void k_hist_cache(const float4* __restrict__ vcache, int n4,
                             const SelState* __restrict__ st, unsigned int* hist) {
  __shared__ unsigned int lh[NBINS];
  const int nb = (LEVEL == 1) ? NBINS : 1024;
  for (int i = threadIdx.x; i < nb; i += BLOCK) lh[i] = 0u;
  __syncthreads();
  const unsigned int prefix = st->prefix;
  const int stride = GRIDB * BLOCK;
  for (int i = blockIdx.x * BLOCK + threadIdx.x; i < n4; i += stride) {
    float4 v4 = vcache[i];
    if (i + stride < n4) __builtin_prefetch((const void*)(vcache + i + stride), 0, 1);
    float vx[4] = {v4.x, v4.y, v4.z, v4.w};
#pragma unroll
    for (int j = 0; j < 4; ++j) {
      unsigned int u = __float_as_uint(vx[j]);
      // sentinel (-1.0f) has the sign bit set -> can never equal a prefix
      if (LEVEL == 1) {
        if ((u >> 21) == prefix) atomicAdd(&lh[(u >> 10) & 0x7FFu], 1u);
      } else {
        if ((u >> 10) == prefix) atomicAdd(&lh[u & 0x3FFu], 1u);
      }
    }
  }
  __syncthreads();
  for (int b = threadIdx.x; b < nb; b += BLOCK) {
    unsigned int h = lh[b];
    if (h) atomicAdd(&hist[b], h);
  }
}

__global__ void k_pass4_cache(const float4* __restrict__ vcache, int n4,
                              const SelState* __restrict__ st, float* partials) {
  __shared__ float lsum[BLOCK / 32];
  const unsigned int T = st->prefix;
  float s = 0.f;
  const int stride = GRIDB * BLOCK;
  for (int i = blockIdx.x * BLOCK + threadIdx.x; i < n4; i += stride) {
    float4 v4 = vcache[i];
    if (i + stride < n4) __builtin_prefetch((const void*)(vcache + i + stride), 0, 1);
    float vx[4] = {v4.x, v4.y, v4.z, v4.w};
#pragma unroll
    for (int j = 0; j < 4; ++j) {
      unsigned int u = __float_as_uint(vx[j]);
      if (!(u >> 31) && u > T) s += vx[j];  // exclude sentinel, strict >
    }
  }
  float wsum = wave_sum_wmma(s);
  int lane = threadIdx.x & 31, wid = threadIdx.x >> 5;
  if (lane == 0) lsum[wid] = wsum;
  __syncthreads();
  if (threadIdx.x == 0) {
    float bs = 0.f;
#pragma unroll
    for (int w = 0; w < BLOCK / 32; ++w) bs += lsum[w];
    partials[blockIdx.x] = bs;
  }
}

__global__ void k_select_refine(const unsigned int* __restrict__ hist, int nbins,
                                int shiftbits, SelState* st, int final_level) {
  if (threadIdx.x != 0 || blockIdx.x != 0) return;
  unsigned int k = st->k, c = st->c;
  unsigned int krem = k - c;  // >= 1 whenever k >= 1; == 0 only when k == 0
  unsigned int acc = 0, chosen = 0, cab = 0;
  bool found = false;
  for (int b = nbins - 1; b >= 0; --b) {
    unsigned int h = hist[b];
    if (acc + h >= krem) { chosen = (unsigned int)b; cab = acc; found = true; break; }
    acc += h;
  }
  if (!found) { chosen = 0; cab = acc; }
  st->prefix = (st->prefix << shiftbits) | chosen;
  st->c = c + cab;
  if (final_level) st->krem = k - st->c;
}

__global__ void k_final(const float* __restrict__ partials, const SelState* __restrict__ st,
                        float* out) {
  __shared__ float red[256];
  float s = 0.f;
  for (int i = threadIdx.x; i < GRIDB; i += 256) s += partials[i];
  red[threadIdx.x] = s;
  __syncthreads();
  for (int off = 128; off > 0; off >>= 1) {
    if (threadIdx.x < off) red[threadIdx.x] += red[threadIdx.x + off];
    __syncthreads();
  }
  if (threadIdx.x == 0) {
    float num = st->pos_sum + red[0];
    unsigned int krem = st->krem;
    if (krem > 0u) num += (float)krem * __uint_as_float(st->prefix);  // ties at T
    float den = (float)(st->pos_count + st->k) + 1e-6f;
    out[0] = num / den;
  }
}

extern "C" void kernel_launch(void* const* d_in, const int* in_sizes, int n_in,
                              void* d_out, int out_size, void* d_ws, size_t ws_size,
                              hipStream_t stream) {
  const float4* pred = (const float4*)d_in[0];
  const float4* gt = (const float4*)d_in[1];
  const float4* mask = (const float4*)d_in[2];
  int n = in_sizes[0];  // 29,491,200 — divisible by 4
  int n4 = n >> 2;

  char* ws = (char*)d_ws;
  SelState* st = (SelState*)ws;
  unsigned int* h0 = (unsigned int*)(ws + 64);
  unsigned int* h1 = (unsigned int*)(ws + 64 + 8192);
  unsigned int* h2 = (unsigned int*)(ws + 64 + 16384);
  float* p1 = (float*)(ws + 20544);
  float* p4 = (float*)(ws + 28736);
  float4* vcache = (float4*)(ws + 40960);

  // Path choice depends only on ws_size (fixed per deployment) -> deterministic.
  const bool use_cache = ws_size >= (size_t)40960 + (size_t)n * 4u;

  // zero state + all histograms (20544 bytes = 5136 u32)
  k_init<<<40, 256, 0, stream>>>((unsigned int*)ws, 5136);

  if (use_cache) {
    k_pass1<1><<<GRIDB, BLOCK, 0, stream>>>(pred, gt, mask, n4, st, h0, p1, vcache);
    k_select0<<<1, 256, 0, stream>>>(p1, st, h0);
    k_hist_cache<1><<<GRIDB, BLOCK, 0, stream>>>(vcache, n4, st, h1);
    k_select_refine<<<1, 1, 0, stream>>>(h1, NBINS, 11, st, 0);
    k_hist_cache<2><<<GRIDB, BLOCK, 0, stream>>>(vcache, n4, st, h2);
    k_select_refine<<<1, 1, 0, stream>>>(h2, 1024, 10, st, 1);
    k_pass4_cache<<<GRIDB, BLOCK, 0, stream>>>(vcache, n4, st, p4);
  } else {
    k_pass1<0><<<GRIDB, BLOCK, 0, stream>>>(pred, gt, mask, n4, st, h0, p1, nullptr);
    k_select0<<<1, 256, 0, stream>>>(p1, st, h0);
    k_hist_pass<1><<<GRIDB, BLOCK, 0, stream>>>(pred, gt, mask, n4, st, h1);
    k_select_refine<<<1, 1, 0, stream>>>(h1, NBINS, 11, st, 0);
    k_hist_pass<2><<<GRIDB, BLOCK, 0, stream>>>(pred, gt, mask, n4, st, h2);
    k_select_refine<<<1, 1, 0, stream>>>(h2, 1024, 10, st, 1);
    k_pass4<<<GRIDB, BLOCK, 0, stream>>>(pred, gt, mask, n4, st, p4);
  }
  k_final<<<1, 256, 0, stream>>>(p4, st, (float*)d_out);
}